// RoIMaskAlignAvg_8538394984627
// MI455X (gfx1250) — compile-verified
//
#include <hip/hip_runtime.h>
#include <stdint.h>

namespace {
constexpr int Bdim = 2, Cdim = 256, Hdim = 200, Wdim = 272, Ndim = 256;
constexpr int AHo = 14, AWo = 14, PHb = 15, PWb = 15, RATIO = 2;
constexpr float FSCALE = 0.25f;
constexpr int WAVES = 8;   // waves per 256-thread block (wave32)
constexpr int CPT   = 2;   // channels per wave
constexpr int OUTF  = CPT * AHo * AWo;   // 392 floats per wave, contiguous in out
}

__global__ __launch_bounds__(256) void roi_align_avg_kernel(
    const float* __restrict__ feat,
    const float* __restrict__ rois,
    float* __restrict__ out)
{
  const int lane = threadIdx.x & 31;
  const int wave = threadIdx.x >> 5;
  const int n    = blockIdx.y;
  const int c0   = (blockIdx.x * WAVES + wave) * CPT;

  // per-wave staging buffer for the 2x196-float output tile (16B aligned:
  // 392 floats = 1568 B = 98 * 16 B, so every wave slice stays aligned)
  __shared__ __align__(16) float obuf[WAVES][OUTF];

  // ---- RoI parameters (uniform -> compiler scalarizes to s_load)
  const float* rp = rois + n * 5;
  const int b = (int)rp[0];
  float x1 = rp[1] * FSCALE, y1 = rp[2] * FSCALE;
  float x2 = rp[3] * FSCALE, y2 = rp[4] * FSCALE;
  {
    // ROI_SCALE=1, SHIFT=0 -> identity, kept for fidelity with reference math
    const float cx = 0.5f * (x1 + x2), cy = 0.5f * (y1 + y2);
    const float hw = 0.5f * (x2 - x1), hh = 0.5f * (y2 - y1);
    x1 = cx - hw; x2 = cx + hw; y1 = cy - hh; y2 = cy + hh;
  }
  const float roi_w = fmaxf(x2 - x1, 1.0f);
  const float roi_h = fmaxf(y2 - y1, 1.0f);
  const float stepx = roi_w / (float)(PWb * RATIO);
  const float stepy = roi_h / (float)(PHb * RATIO);

  const int il = (lane < 2 * PWb - 1) ? lane : (2 * PWb - 1); // clamp lanes 30,31

  // ---- per-lane horizontal taps (lane == ix), validity folded into weights
  const float sx  = x1 + ((float)il + 0.5f) * stepx;
  const float vx  = (sx > -1.0f && sx < (float)Wdim) ? 1.0f : 0.0f;
  const float cxs = fminf(fmaxf(sx, 0.0f), (float)(Wdim - 1));
  const float xlof = floorf(cxs);
  const int   xlo  = (int)xlof;
  const int   xhi  = (xlo + 1 < Wdim) ? xlo + 1 : Wdim - 1;
  const float fx   = cxs - xlof;
  const float wx0  = (1.0f - fx) * vx;
  const float wx1  = fx * vx;

  // ---- per-lane vertical taps for sample row iy == lane (broadcast via shfl)
  const float sy  = y1 + ((float)il + 0.5f) * stepy;
  const float vy  = (sy > -1.0f && sy < (float)Hdim) ? 1.0f : 0.0f;
  const float cys = fminf(fmaxf(sy, 0.0f), (float)(Hdim - 1));
  const float ylof = floorf(cys);
  const int   ylo_l = (int)ylof;
  const int   yhi_l = (ylo_l + 1 < Hdim) ? ylo_l + 1 : Hdim - 1;
  const float fy    = cys - ylof;
  const float wy0_l = (1.0f - fy) * vy;
  const float wy1_l = fy * vy;

  const float* fb[CPT];
#pragma unroll
  for (int q = 0; q < CPT; ++q)
    fb[q] = feat + ((size_t)b * Cdim + (size_t)(c0 + q)) * (size_t)(Hdim * Wdim);

  // CDNA5 speculative prefetch (global_prefetch_b8) to warm WGP$/L2
  __builtin_prefetch(fb[0]       + (size_t)ylo_l * Wdim + xlo, 0, 3);
  __builtin_prefetch(fb[CPT - 1] + (size_t)yhi_l * Wdim + xlo, 0, 3);

  float prev[CPT];
#pragma unroll
  for (int q = 0; q < CPT; ++q) prev[q] = 0.0f;

  for (int k = 0; k < PHb; ++k) {
    float cur[CPT];
#pragma unroll
    for (int q = 0; q < CPT; ++q) cur[q] = 0.0f;

#pragma unroll
    for (int t = 0; t < 2; ++t) {
      const int   iy  = 2 * k + t;
      const int   ylo = __shfl(ylo_l, iy, 32);
      const int   yhi = __shfl(yhi_l, iy, 32);
      const float wy0 = __shfl(wy0_l, iy, 32);
      const float wy1 = __shfl(wy1_l, iy, 32);
#pragma unroll
      for (int q = 0; q < CPT; ++q) {
        const float* r0 = fb[q] + (size_t)ylo * Wdim;
        const float* r1 = fb[q] + (size_t)yhi * Wdim;
        const float f00 = r0[xlo], f01 = r0[xhi];
        const float f10 = r1[xlo], f11 = r1[xhi];
        cur[q] += wy0 * (wx0 * f00 + wx1 * f01)
                + wy1 * (wx0 * f10 + wx1 * f11);
      }
    }

    if (k >= 1) {
#pragma unroll
      for (int q = 0; q < CPT; ++q) {
        // vertical 4-row window (rows 2(k-1)..2k+1), then horizontal 4-col window
        float v = prev[q] + cur[q];
        float u = v + __shfl_xor(v, 1, 32);   // u[2j] = p[j]   (col pair sums)
        float w = u + __shfl_down(u, 2, 32);  // w[2j] = p[j]+p[j+1]
        float o = __shfl(w, lane * 2, 32);    // compact: lane j <- w[2j]
        if (lane < AWo) {
          obuf[wave][q * (AHo * AWo) + (k - 1) * AWo + lane] = o * (1.0f / 16.0f);
        }
      }
    }
#pragma unroll
    for (int q = 0; q < CPT; ++q) prev[q] = cur[q];
  }

  // ---- flush the per-wave tile LDS -> global with CDNA5 async DMA stores.
  // 392 floats are contiguous in out at a 16B-aligned base (784 B per channel).
  asm volatile("s_wait_dscnt 0x0" ::: "memory");  // LDS writes visible to async engine
  const float* gbase = out + ((size_t)n * Cdim + (size_t)c0) * (size_t)(AHo * AWo);
  const uint32_t lbase = (uint32_t)(uintptr_t)(&obuf[wave][0]);
#pragma unroll
  for (int i = 0; i < 4; ++i) {
    const int chunk = i * 32 + lane;          // 16-byte chunks, 98 total
    if (chunk < OUTF / 4) {
      const float* gp = gbase + chunk * 4;
      const uint32_t lp = lbase + (uint32_t)chunk * 16u;
      asm volatile("global_store_async_from_lds_b128 %0, %1, off"
                   :: "v"(gp), "v"(lp)
                   : "memory");
    }
  }
  asm volatile("s_wait_asynccnt 0x0" ::: "memory");
}

extern "C" void kernel_launch(void* const* d_in, const int* in_sizes, int n_in,
                              void* d_out, int out_size, void* d_ws, size_t ws_size,
                              hipStream_t stream) {
  (void)in_sizes; (void)n_in; (void)out_size; (void)d_ws; (void)ws_size;
  const float* feat = (const float*)d_in[0];
  const float* rois = (const float*)d_in[1];
  float* out = (float*)d_out;

  dim3 grid(Cdim / (WAVES * CPT), Ndim);  // (16, 256) blocks, 256 threads = 8 wave32
  roi_align_avg_kernel<<<grid, dim3(256), 0, stream>>>(feat, rois, out);
}